// AffNet_77077483094228
// MI455X (gfx1250) — compile-verified
//
#include <hip/hip_runtime.h>
#include <math.h>

typedef __attribute__((ext_vector_type(2))) float v2f;
typedef __attribute__((ext_vector_type(8))) float v8f;

#define N_NODES 8192
#define N_FEAT  512
#define EMB     128
#define N_HEADS 4
#define EPS     1e-6f

#define BT  64    // output block tile edge (per side)
#define LDP 132   // padded LDS row stride (floats): 132 % 64 = 4 -> conflict-free
#define SST 65    // S exchange tile stride

#if defined(__has_builtin)
#if __has_builtin(__builtin_amdgcn_global_load_async_to_lds_b128)
#define HAS_ASYNC_LDS 1
#endif
#endif

// 16-byte global -> LDS copy; async (ASYNCcnt-tracked, no VGPR data) if available
__device__ __forceinline__ void copy16_to_lds(float* lds_dst, const float* gsrc) {
#ifdef HAS_ASYNC_LDS
  typedef int v4i_ __attribute__((__vector_size__(16)));
  typedef __attribute__((address_space(1))) v4i_* gp_t;
  typedef __attribute__((address_space(3))) v4i_* lp_t;
  __builtin_amdgcn_global_load_async_to_lds_b128((gp_t)(gsrc), (lp_t)(lds_dst),
                                                 0, 0);
#else
  *reinterpret_cast<float4*>(lds_dst) = *reinterpret_cast<const float4*>(gsrc);
#endif
}

__device__ __forceinline__ void wait_async_and_barrier() {
#ifdef HAS_ASYNC_LDS
#if __has_builtin(__builtin_amdgcn_s_wait_asynccnt)
  __builtin_amdgcn_s_wait_asynccnt(0);
#else
  asm volatile("s_wait_asynccnt 0x0" ::: "memory");
#endif
#endif
  __syncthreads();
}

// stage a 64x128 f32 tile (row base gsrc) into padded LDS, 128 threads
__device__ __forceinline__ void stage_tile_128(float* lds, const float* gsrc,
                                               int tid) {
#pragma unroll
  for (int it = 0; it < 16; ++it) {
    const int idx = it * 128 + tid;
    const int r  = idx >> 5;
    const int c4 = (idx & 31) * 4;
    copy16_to_lds(&lds[r * LDP + c4], &gsrc[(size_t)r * EMB + c4]);
  }
}

// --------------------------------------------------------------------------
// Kernel 1: X_emb = X @ W  via V_WMMA_F32_16X16X4_F32, one wave per 16x16 tile
// --------------------------------------------------------------------------
__global__ __launch_bounds__(32) void emb_gemm_kernel(const float* __restrict__ X,
                                                      const float* __restrict__ W,
                                                      float* __restrict__ Xe) {
  const int bj   = blockIdx.x;          // EMB/16 tiles
  const int bi   = blockIdx.y;          // N/16 tiles
  const int lane = threadIdx.x & 31;
  const int hl   = lane >> 4;
  const int l16  = lane & 15;
  const int row  = bi * 16 + l16;
  const int col  = bj * 16 + l16;

  v8f acc = {};
  for (int kb = 0; kb < N_FEAT; kb += 4) {
    const int k = kb + 2 * hl;
    v2f a = *reinterpret_cast<const v2f*>(&X[(size_t)row * N_FEAT + k]);
    v2f b;
    b.x = W[(size_t)k * EMB + col];
    b.y = W[(size_t)(k + 1) * EMB + col];
    acc = __builtin_amdgcn_wmma_f32_16x16x4_f32(false, a, false, b, (short)0,
                                                acc, false, false);
  }
#pragma unroll
  for (int v = 0; v < 8; ++v) {
    const int m = bi * 16 + v + 8 * hl;
    Xe[(size_t)m * EMB + col] = acc[v];
  }
}

// --------------------------------------------------------------------------
// Kernel 2: row L2 norms of X_emb [8192x128] and Z [4*8192 x 128]
// --------------------------------------------------------------------------
__global__ __launch_bounds__(128) void norm_kernel(const float* __restrict__ Xe,
                                                   const float* __restrict__ Z,
                                                   float* __restrict__ Xn,
                                                   float* __restrict__ Zn) {
  __shared__ float red[128];
  const int r = blockIdx.x;
  const int t = threadIdx.x;
  const float* src;
  float* dst;
  if (r < N_NODES) {
    src = Xe + (size_t)r * EMB;
    dst = Xn + r;
  } else {
    const int rr = r - N_NODES;
    src = Z + (size_t)rr * EMB;
    dst = Zn + rr;
  }
  const float v = src[t];
  red[t] = v * v;
  __syncthreads();
  for (int s = 64; s > 0; s >>= 1) {
    if (t < s) red[t] += red[t + s];
    __syncthreads();
  }
  if (t == 0) *dst = sqrtf(red[0]);
}

// --------------------------------------------------------------------------
// Kernel 3: tile-pair affinity kernel, 128 threads = 4 waves.
// Waves 0-1 compute S_IJ (64x64), waves 2-3 compute S_JI; each wave owns a
// 32x64 strip -> per k-step: 2 A-frags + 4 B-frags = 6 ds loads, 8 WMMAs.
// Exchange via LDS, symmetrize, affine, pow(beta), write both mirror tiles.
// --------------------------------------------------------------------------
__global__ __launch_bounds__(128) void aff_kernel(const float* __restrict__ Z,
                                                  const float* __restrict__ Xe,
                                                  const float* __restrict__ Xn,
                                                  const float* __restrict__ Zn,
                                                  const float* __restrict__ beta_p,
                                                  float* __restrict__ out) {
  const int bi = blockIdx.y;
  const int bj = blockIdx.x;
  if (bi > bj) return;
  const int ibase = bi * BT;
  const int jbase = bj * BT;

  const int tid  = threadIdx.x;
  const int wave = tid >> 5;
  const int lane = tid & 31;
  const int hl   = lane >> 4;
  const int l16  = lane & 15;
  const int side = wave >> 1;   // 0: (I,J) tile, 1: (J,I) tile
  const int wr   = wave & 1;    // 32-row half of the 64x64 tile

  __shared__ float smem[4 * BT * LDP + 2 * BT + 2 * N_HEADS * BT];
  float* XeI = smem;                   // 64 x LDP
  float* XeJ = XeI + BT * LDP;         // 64 x LDP
  float* ZA  = XeJ + BT * LDP;         // 64 x LDP  (Z rows for I)
  float* ZB  = ZA + BT * LDP;          // 64 x LDP  (Z rows for J)
  float* XnI = ZB + BT * LDP;          // 64
  float* XnJ = XnI + BT;               // 64
  float* ZnI = XnJ + BT;               // 4 x 64
  float* ZnJ = ZnI + N_HEADS * BT;     // 4 x 64
  float* SIJ = ZA;                     // reused after head loop (64 x SST)
  float* SJI = ZA + BT * SST;          // fits within ZA region (8320 <= 8448)

  // ---- stage X_emb tiles (async) + norms ----
  stage_tile_128(XeI, &Xe[(size_t)ibase * EMB], tid);
  stage_tile_128(XeJ, &Xe[(size_t)jbase * EMB], tid);
  if (tid < BT) {
    XnI[tid] = Xn[ibase + tid];
    XnJ[tid] = Xn[jbase + tid];
  }
  for (int q = tid; q < N_HEADS * BT; q += 128) {
    const int h = q >> 6;
    const int r = q & 63;
    ZnI[q] = Zn[h * N_NODES + ibase + r];
    ZnJ[q] = Zn[h * N_NODES + jbase + r];
  }

  float best[2][4][8];
#pragma unroll
  for (int fr = 0; fr < 2; ++fr)
#pragma unroll
    for (int fc = 0; fc < 4; ++fc)
#pragma unroll
      for (int v = 0; v < 8; ++v) best[fr][fc][v] = -3.0e38f;

  const int r0 = wr * 32 + l16;   // A-fragment rows within tile
  const int r1 = r0 + 16;

  for (int h = 0; h < N_HEADS; ++h) {
    __syncthreads();   // previous-iteration LDS reads done before restaging
    stage_tile_128(ZA, &Z[((size_t)h * N_NODES + ibase) * EMB], tid);
    stage_tile_128(ZB, &Z[((size_t)h * N_NODES + jbase) * EMB], tid);
    if (h + 1 < N_HEADS) {
      // warm L2/WGP$ for the next head while this one computes
      const int r = tid >> 1;
      const int c = (tid & 1) * 64;
      __builtin_prefetch(&Z[((size_t)(h + 1) * N_NODES + ibase + r) * EMB + c], 0, 1);
      __builtin_prefetch(&Z[((size_t)(h + 1) * N_NODES + jbase + r) * EMB + c], 0, 1);
    }
    wait_async_and_barrier();

    const float* Zt = side ? ZB : ZA;   // A rows (Z) for this side
    const float* Xo = side ? XeI : XeJ; // B cols (X_emb of the other side)

    v8f acc[2][4] = {};
#pragma unroll 4
    for (int kb = 0; kb < EMB; kb += 4) {
      const int ka = kb + 2 * hl;
      v2f a0 = *reinterpret_cast<const v2f*>(&Zt[r0 * LDP + ka]);
      v2f a1 = *reinterpret_cast<const v2f*>(&Zt[r1 * LDP + ka]);
#pragma unroll
      for (int fc = 0; fc < 4; ++fc) {
        v2f b = *reinterpret_cast<const v2f*>(&Xo[(fc * 16 + l16) * LDP + ka]);
        acc[0][fc] = __builtin_amdgcn_wmma_f32_16x16x4_f32(false, a0, false, b,
                                                           (short)0, acc[0][fc],
                                                           false, false);
        acc[1][fc] = __builtin_amdgcn_wmma_f32_16x16x4_f32(false, a1, false, b,
                                                           (short)0, acc[1][fc],
                                                           false, false);
      }
    }

    // ---- cosine normalization + running max over heads ----
    const float* Xnrm = side ? XnI : XnJ;
    const float* Znrm = side ? ZnJ : ZnI;
#pragma unroll
    for (int fr = 0; fr < 2; ++fr)
#pragma unroll
      for (int fc = 0; fc < 4; ++fc) {
        const float xn = Xnrm[fc * 16 + l16] + EPS;
#pragma unroll
        for (int v = 0; v < 8; ++v) {
          const int mt = wr * 32 + fr * 16 + v + 8 * hl;
          const float zn = Znrm[h * BT + mt] + EPS;
          const float s = acc[fr][fc][v] / (xn * zn);
          best[fr][fc][v] = fmaxf(best[fr][fc][v], s);
        }
      }
  }

  // ---- exchange maxed tiles through LDS ----
  __syncthreads();
  float* S = side ? SJI : SIJ;
#pragma unroll
  for (int fr = 0; fr < 2; ++fr)
#pragma unroll
    for (int fc = 0; fc < 4; ++fc)
#pragma unroll
      for (int v = 0; v < 8; ++v)
        S[(wr * 32 + fr * 16 + v + 8 * hl) * SST + fc * 16 + l16] =
            best[fr][fc][v];
  __syncthreads();

  // ---- symmetrize + affine + pow, write both mirror tiles (coalesced) ----
  const float beta  = beta_p[0];
  const bool  dopow = (beta != 1.0f);   // wave-uniform
#pragma unroll 2
  for (int t = 0; t < 32; ++t) {
    const int idx = t * 128 + tid;
    const int a = idx >> 6;
    const int b = idx & 63;
    float v = 0.5f * (SIJ[a * SST + b] + SJI[b * SST + a]);
    v = 0.5f * (v + 1.0f);
    if (dopow) v = __powf(v, beta);
    out[(size_t)(ibase + a) * N_NODES + (jbase + b)] = v;
  }
  if (bi != bj) {
#pragma unroll 2
    for (int t = 0; t < 32; ++t) {
      const int idx = t * 128 + tid;
      const int b = idx >> 6;   // row within J tile
      const int a = idx & 63;   // col within I tile (contiguous -> coalesced)
      float v = 0.5f * (SIJ[a * SST + b] + SJI[b * SST + a]);
      v = 0.5f * (v + 1.0f);
      if (dopow) v = __powf(v, beta);
      out[(size_t)(jbase + b) * N_NODES + (ibase + a)] = v;
    }
  }
}

// --------------------------------------------------------------------------
extern "C" void kernel_launch(void* const* d_in, const int* in_sizes, int n_in,
                              void* d_out, int out_size, void* d_ws, size_t ws_size,
                              hipStream_t stream) {
  const float* X    = (const float*)d_in[0];   // [8192, 512]
  const float* W    = (const float*)d_in[1];   // [512, 128]
  const float* Z    = (const float*)d_in[2];   // [4, 8192, 128]
  const float* beta = (const float*)d_in[3];   // scalar
  float* out = (float*)d_out;                  // [8192, 8192]

  float* Xe = (float*)d_ws;                    // [8192, 128]
  float* Xn = Xe + (size_t)N_NODES * EMB;      // [8192]
  float* Zn = Xn + N_NODES;                    // [4, 8192]

  emb_gemm_kernel<<<dim3(EMB / 16, N_NODES / 16), 32, 0, stream>>>(X, W, Xe);
  norm_kernel<<<N_NODES + N_HEADS * N_NODES, 128, 0, stream>>>(Xe, Z, Xn, Zn);
  aff_kernel<<<dim3(N_NODES / BT, N_NODES / BT), 128, 0, stream>>>(Z, Xe, Xn, Zn,
                                                                   beta, out);
}